// Ani_78073915506935
// MI455X (gfx1250) — compile-verified
//
#include <hip/hip_runtime.h>
#include <hip/hip_bf16.h>
#include <math.h>

typedef __attribute__((ext_vector_type(2))) float v2f;
typedef __attribute__((ext_vector_type(8))) float v8f;

#define NMAXPAD 256   // supports up to 256 atoms (harness uses 200)

__device__ __forceinline__ float fcut(float d, float rc) {
  return (d <= rc) ? (0.5f * __cosf(3.14159265358979f * d / rc) + 0.5f) : 0.0f;
}

__global__ __launch_bounds__(256) void aev_kernel(const float* __restrict__ coords,
                                                  float* __restrict__ out, int n) {
  const int i    = blockIdx.x;
  const int tid  = threadIdx.x;
  const int lane = tid & 31;
  const int wave = tid >> 5;
  const int NT   = (n + 15) >> 4;   // 16-wide tiles in j/k
  const int NPAD = NT << 4;

  __shared__ float4 s_unit[NMAXPAD];   // unit vectors, 4th comp = 0 (K-pad for WMMA)
  __shared__ float  s_d[NMAXPAD];
  __shared__ float  s_fca[NMAXPAD];
  __shared__ float  s_rad[16];
  __shared__ float  s_acc[32];

  if (tid < 16) s_rad[tid] = 0.0f;
  if (tid < 32) s_acc[tid] = 0.0f;
  __syncthreads();

  const float SHFR[16] = {0.9f, 1.16875f, 1.4375f, 1.70625f, 1.975f, 2.24375f,
                          2.5125f, 2.78125f, 3.05f, 3.31875f, 3.5875f, 3.85625f,
                          4.125f, 4.39375f, 4.6625f, 4.93125f};

  // ---------------- phase 1: pair quantities + radial subAEV ----------------
  const float cix = coords[3*i+0], ciy = coords[3*i+1], ciz = coords[3*i+2];
  {
    const int j = tid;
    float d = 1.0f, fa = 0.0f, fr = 0.0f, ux = 0.0f, uy = 0.0f, uz = 0.0f;
    if (j < n && j != i) {
      float dx = cix - coords[3*j+0];
      float dy = ciy - coords[3*j+1];
      float dz = ciz - coords[3*j+2];
      float d2 = dx*dx + dy*dy + dz*dz;
      d = __fsqrt_rn(d2);
      float inv = __frcp_rn(d);
      ux = dx*inv; uy = dy*inv; uz = dz*inv;
      fr = fcut(d, 5.2f);   // RCR
      fa = fcut(d, 3.5f);   // RCA
    }
    if (j < NPAD) {
      s_unit[j] = make_float4(ux, uy, uz, 0.0f);
      s_d[j]    = d;
      s_fca[j]  = fa;
    }
    // radial: 0.25*exp(-16*(d-ShfR)^2)*fc_r, fr==0 kills invalid/diag/pad lanes
    #pragma unroll
    for (int s = 0; s < 16; ++s) {
      float t = d - SHFR[s];
      float v = 0.25f * __expf(-16.0f * t * t) * fr;
      v += __shfl_xor(v, 16, 32);
      v += __shfl_xor(v,  8, 32);
      v += __shfl_xor(v,  4, 32);
      v += __shfl_xor(v,  2, 32);
      v += __shfl_xor(v,  1, 32);
      if (lane == 0) atomicAdd(&s_rad[s], v);
    }
  }
  __syncthreads();

  // ---------------- phase 2: angular subAEV via WMMA Gram tiles ----------------
  // ShfZ = (2z+1)*pi/16  ->  hard-coded cos/sin
  const float CZ[8] = { 0.98078528f,  0.83146961f,  0.55557023f,  0.19509032f,
                       -0.19509032f, -0.55557023f, -0.83146961f, -0.98078528f};
  const float SZ[8] = { 0.19509032f,  0.55557023f,  0.83146961f,  0.98078528f,
                        0.98078528f,  0.83146961f,  0.55557023f,  0.19509032f};
  const float SHFA[4] = {0.9f, 1.55f, 2.2f, 2.85f};

  float acc[32];
  #pragma unroll
  for (int m = 0; m < 32; ++m) acc[m] = 0.0f;

  const int half = lane >> 4;   // WMMA 32-bit operand layout: lanes 16-31 hold K=2,3 / M+8
  const int nloc = lane & 15;

  int u = 0;
  for (int jt = 0; jt < NT; ++jt) {
    for (int kt = jt; kt < NT; ++kt, ++u) {
      if ((u & 7) != wave) continue;          // wave-uniform tile ownership (8 waves)
      const int   jbase = jt << 4, kbase = kt << 4;
      const float wgt   = (jt == kt) ? 1.0f : 2.0f;   // j<->k symmetry

      // A: 16x4 rows of unit_j ; B: 4x16 cols of unit_k  (K padded with 0)
      const float* ap = (const float*)&s_unit[jbase + nloc];
      const float* bp = (const float*)&s_unit[kbase + nloc];
      v2f A, B;
      A.x = ap[2*half]; A.y = ap[2*half + 1];
      B.x = bp[2*half]; B.y = bp[2*half + 1];
      v8f C = {0.f,0.f,0.f,0.f,0.f,0.f,0.f,0.f};
      v8f D = __builtin_amdgcn_wmma_f32_16x16x4_f32(false, A, false, B,
                                                    (short)0, C, false, false);

      const int   kg = kbase + nloc;          // N = lane&15
      const float dk = s_d[kg];
      const float fk = s_fca[kg] * wgt;
      #pragma unroll
      for (int v = 0; v < 8; ++v) {
        const int   jg = jbase + v + 8*half;  // M = vgpr + 8*(lane>=16)
        const float dj = s_d[jg];
        float w = s_fca[jg] * fk;
        if (jg == kg) w = 0.0f;               // exclude j==k
        float raw = fminf(1.0f, fmaxf(-1.0f, D[v]));
        float ct  = 0.95f * raw;              // cos(theta)
        // 1-ct^2 in [0.0975, 1]: never denormal/negative -> raw v_sqrt_f32 is safe
        float st  = __builtin_amdgcn_sqrtf(1.0f - ct*ct);  // sin(theta)
        float davg = 0.5f * (dj + dk);
        float f2[4];
        #pragma unroll
        for (int a = 0; a < 4; ++a) {
          float t = davg - SHFA[a];
          f2[a] = w * __expf(-8.0f * t * t);
        }
        #pragma unroll
        for (int z = 0; z < 8; ++z) {
          float base = 0.5f + 0.5f * (ct*CZ[z] + st*SZ[z]);   // (1+cos(theta-ShfZ))/2
          float p = base*base; p *= p; p *= p; p *= p; p *= p; // ^32
          acc[0*8+z] += f2[0]*p;
          acc[1*8+z] += f2[1]*p;
          acc[2*8+z] += f2[2]*p;
          acc[3*8+z] += f2[3]*p;
        }
      }
    }
  }

  #pragma unroll
  for (int m = 0; m < 32; ++m) {
    float v = acc[m];
    v += __shfl_xor(v, 16, 32);
    v += __shfl_xor(v,  8, 32);
    v += __shfl_xor(v,  4, 32);
    v += __shfl_xor(v,  2, 32);
    v += __shfl_xor(v,  1, 32);
    if (lane == 0) atomicAdd(&s_acc[m], v);
  }
  __syncthreads();

  // ---------------- write row i: [0:16] radial, [64:96] angular, rest zero ----------------
  for (int idx = tid; idx < 384; idx += 256) {
    float val = 0.0f;
    if (idx < 16)                   val = s_rad[idx];
    else if (idx >= 64 && idx < 96) val = s_acc[idx - 64];
    out[i * 384 + idx] = val;
  }
}

extern "C" void kernel_launch(void* const* d_in, const int* in_sizes, int n_in,
                              void* d_out, int out_size, void* d_ws, size_t ws_size,
                              hipStream_t stream) {
  const float* coords = (const float*)d_in[0];
  float* out = (float*)d_out;
  int n = in_sizes[0] / 3;          // coordinates are (n, 3) f32
  if (n > NMAXPAD) n = NMAXPAD;
  if (n <= 0) return;
  aev_kernel<<<n, 256, 0, stream>>>(coords, out, n);
  (void)d_ws; (void)ws_size; (void)n_in; (void)out_size;
}